// Encoder_85899345920647
// MI455X (gfx1250) — compile-verified
//
#include <hip/hip_runtime.h>
#include <hip/hip_bf16.h>

// Problem constants (from reference)
#define B_   32
#define N_   512
#define DIN  256
#define DSC  32
#define C_   48
#define H_   152
#define W_   160
#define HW_  (H_ * W_)
#define CTOT (C_ + DSC)          // 80 output channels
#define NROWS (B_ * N_)          // 16384 rows in the GEMM

typedef float v2f __attribute__((ext_vector_type(2)));
typedef float v8f __attribute__((ext_vector_type(8)));

// ---------------------------------------------------------------------------
// Kernel 1: proj[16384, 32] = relu(E[16384,256] @ Wp[256,32] + b) * mask
// One wave computes a 16x32 output tile using V_WMMA_F32_16X16X4_F32.
//
// A-layout (ISA 7.12.2, 32-bit A 16x4): lanes 0-15 hold K = {k, k+1},
// lanes 16-31 hold K = {k+2, k+3}; M = lane % 16.
// B-layout (rows striped across lanes per VGPR): VGPR v, lanes 0-15 -> K = k+v,
// lanes 16-31 -> K = k+v+2; N = lane % 16.
// C/D layout: VGPR v -> row m0 + v + 8*(lane>=16), col = lane % 16.
//
// W_proj is staged into LDS *pair-interleaved*:
//   sW[(k/2)*64 + col*2 + (k&1)] = Wp[k*DSC + col]
// so each lane's B-fragment {W[kk][col], W[kk+1][col]} is one aligned 8-byte
// LDS word -> single ds_load_b64, no register repacking before the WMMA.
// ---------------------------------------------------------------------------
__global__ __launch_bounds__(256) void gemm_proj_kernel(
    const float* __restrict__ E,     // [NROWS, DIN]
    const float* __restrict__ Wp,    // [DIN, DSC]
    const float* __restrict__ bias,  // [DSC]
    const float* __restrict__ mask,  // [NROWS]
    float* __restrict__ proj)        // [NROWS, DSC]
{
    __shared__ float sW[DIN * DSC];  // 32 KB, pair-interleaved layout

    const int tid = threadIdx.x;
    // Stage + interleave W_proj (8192 elements / 256 threads = 32 each).
    #pragma unroll
    for (int i = 0; i < (DIN * DSC) / 256; ++i) {
        const int idx = tid + i * 256;        // linear index into Wp
        const int k = idx >> 5;               // K row   (0..255)
        const int c = idx & 31;               // column  (0..31)
        sW[(k >> 1) * 64 + c * 2 + (k & 1)] = Wp[idx];
    }
    __syncthreads();

    const int wave = tid >> 5;               // 0..7
    const int lane = tid & 31;
    const int half = lane >> 4;              // 0/1: K-pair select
    const int l16  = lane & 15;              // M / N within tile
    const int m0   = (blockIdx.x * 8 + wave) * 16;   // row base of this tile

    const float* __restrict__ Arow = E + (size_t)(m0 + l16) * DIN + half * 2;
    // v2f view of interleaved LDS: B-fragment for pair p, column c sits at
    // v2f index p*32 + c. For this lane: p = k/2 + half, c0 = l16, c1 = 16+l16.
    const v2f* __restrict__ sB = ((const v2f*)sW) + half * 32 + l16;

    v8f acc0 = {};   // cols 0..15
    v8f acc1 = {};   // cols 16..31

    #pragma unroll 4
    for (int k = 0; k < DIN; k += 4) {
        v2f a;
        a.x = Arow[k];
        a.y = Arow[k + 1];

        const v2f* bp = sB + (k >> 1) * 32;
        const v2f b0 = bp[0];       // ds_load_b64
        const v2f b1 = bp[16];      // ds_load_b64 (128 B apart -> 2addr fuse)

        // D = A * B + C   (fp32 WMMA, 8-arg form)
        acc0 = __builtin_amdgcn_wmma_f32_16x16x4_f32(
            false, a, false, b0, (short)0, acc0, false, false);
        acc1 = __builtin_amdgcn_wmma_f32_16x16x4_f32(
            false, a, false, b1, (short)0, acc1, false, false);
    }

    // Epilogue: bias + ReLU + mask, store per C/D VGPR layout.
    const float bias0 = bias[l16];
    const float bias1 = bias[16 + l16];
    #pragma unroll
    for (int v = 0; v < 8; ++v) {
        const int row = m0 + v + half * 8;
        const float mk = mask[row];
        float r0 = acc0[v] + bias0;
        float r1 = acc1[v] + bias1;
        r0 = (r0 > 0.0f ? r0 : 0.0f) * mk;
        r1 = (r1 > 0.0f ? r1 : 0.0f) * mk;
        proj[(size_t)row * DSC + l16]      = r0;
        proj[(size_t)row * DSC + 16 + l16] = r1;
    }
}

// ---------------------------------------------------------------------------
// Kernel 2: out[b, 0:48, :, :] = spatial_info ; out[b, 48:80, :, :] = 0
// float4-vectorized (HW = 24320 divisible by 4). 249 MB of stores — this
// kernel dominates runtime and is pure streaming bandwidth.
// ---------------------------------------------------------------------------
__global__ __launch_bounds__(256) void concat_zero_kernel(
    const float* __restrict__ spatial,   // [B, C, H, W]
    float* __restrict__ out)             // [B, CTOT, H, W]
{
    const long long i4 = (long long)blockIdx.x * blockDim.x + threadIdx.x;
    const int P4 = HW_ / 4;                       // 6080 float4 per channel
    const long long total4 = (long long)B_ * CTOT * P4;
    if (i4 >= total4) return;

    const int p4 = (int)(i4 % P4);
    const int c  = (int)((i4 / P4) % CTOT);
    const int b  = (int)(i4 / ((long long)P4 * CTOT));

    float4 v;
    if (c < C_) {
        v = ((const float4*)spatial)[((size_t)b * C_ + c) * P4 + p4];
    } else {
        v = make_float4(0.0f, 0.0f, 0.0f, 0.0f);
    }
    ((float4*)out)[i4] = v;
}

// ---------------------------------------------------------------------------
// Kernel 3: scatter-add. One lane per (entity, channel):
//   out[b, 48+d, clamp(y), clamp(x)] += proj[e, d]
// proj loads are fully coalesced (32 consecutive floats per wave).
// ---------------------------------------------------------------------------
__global__ __launch_bounds__(256) void scatter_kernel(
    const float* __restrict__ proj,   // [NROWS, DSC]
    const int*   __restrict__ loc,    // [B, N, 2] (y, x)
    float* __restrict__ out)          // [B, CTOT, H, W]
{
    const int t = blockIdx.x * 256 + threadIdx.x;
    const int e = t >> 5;            // entity index 0..16383
    const int d = t & 31;            // channel 0..31
    if (e >= NROWS) return;

    int y = loc[2 * e + 0];
    int x = loc[2 * e + 1];
    y = min(max(y, 0), H_ - 1);
    x = min(max(x, 0), W_ - 1);

    const int b = e >> 9;            // e / N_
    const float v = proj[(size_t)e * DSC + d];
    const size_t oidx = (((size_t)b * CTOT + C_ + d) * H_ + y) * W_ + x;
    atomicAdd(&out[oidx], v);
}

// ---------------------------------------------------------------------------
extern "C" void kernel_launch(void* const* d_in, const int* in_sizes, int n_in,
                              void* d_out, int out_size, void* d_ws, size_t ws_size,
                              hipStream_t stream) {
    const float* spatial = (const float*)d_in[0];   // [B, C, H, W]
    const float* E       = (const float*)d_in[1];   // [B, N, DIN]
    const float* mask    = (const float*)d_in[2];   // [B, N]
    const int*   loc     = (const int*)  d_in[3];   // [B, N, 2]
    const float* Wp      = (const float*)d_in[4];   // [DIN, DSC]
    const float* bp      = (const float*)d_in[5];   // [DSC]
    float* out  = (float*)d_out;                    // [B, CTOT, H, W]
    float* proj = (float*)d_ws;                     // [NROWS, DSC] = 2 MB scratch

    // 1) WMMA GEMM: 1024 tiles of 16 rows, 8 waves/block -> 128 blocks.
    gemm_proj_kernel<<<NROWS / (16 * 8), 256, 0, stream>>>(E, Wp, bp, mask, proj);

    // 2) Copy spatial channels + zero scatter channels (float4 granularity).
    const long long total4 = (long long)B_ * CTOT * (HW_ / 4);
    const int cz_blocks = (int)((total4 + 255) / 256);
    concat_zero_kernel<<<cz_blocks, 256, 0, stream>>>(spatial, out);

    // 3) Scatter-add (depends on 1 and 2; stream order serializes).
    scatter_kernel<<<(NROWS * DSC) / 256, 256, 0, stream>>>(proj, loc, out);
}